// TreecrfLoss_mcse_44659069943948
// MI455X (gfx1250) — compile-verified
//
#include <hip/hip_runtime.h>

// TreeCRF span loss for MI455X (gfx1250, wave32).
// Inside/outside DP is LDS-resident: one workgroup per batch element, chart in
// one padded LDS array (stride 129 words -> bank-conflict-free column reads):
// beta in the upper triangle, outside-F in the lower triangle.
// Each width splits every cell's reduction over R=2..32 lanes (shfl_xor merge),
// online logsumexp uses a single v_exp_f32 per term.
// BCE reductions stream through CDNA5 async global->LDS copies and are reduced
// with V_WMMA_F32_16X16X4_F32 (ones-matrix reduction, exact f32).

#define LDIM 128
#define STR 129            // padded LDS row stride (129 % 64 == 1)
#define BDIM 32
#define NEGV (-1.0e9f)
#define LOGEPS (-87.49823353f)   // ln(1e-38)

typedef float v2f __attribute__((ext_vector_type(2)));
typedef float v8f __attribute__((ext_vector_type(8)));

#if defined(__has_builtin)
#if __has_builtin(__builtin_amdgcn_wmma_f32_16x16x4_f32)
#define HAVE_WMMA_X4 1
#endif
#endif

__device__ __forceinline__ float wave_sum(float v) {
  for (int o = 16; o > 0; o >>= 1) v += __shfl_down(v, o, 32);
  return v;
}

// online logsumexp accumulate: one transcendental per term
__device__ __forceinline__ void lse_acc(float& m, float& s, float t) {
  float d = t - m;
  float e = __expf(-fabsf(d));
  bool gt = d > 0.0f;
  s = gt ? __fmaf_rn(s, e, 1.0f) : (s + e);
  m = gt ? t : m;
}

// merge two (max, sum) partials: one transcendental
__device__ __forceinline__ void lse_combine(float& m, float& s, float om, float os) {
  float d = om - m;
  float e = __expf(-fabsf(d));
  s = (d > 0.0f) ? __fmaf_rn(s, e, os) : __fmaf_rn(os, e, s);
  m = fmaxf(m, om);
}

__device__ __forceinline__ float lse2(float a, float b) {
  float mm = fmaxf(a, b);
  return mm + __logf(1.0f + __expf(-fabsf(a - b)));
}

__global__ __launch_bounds__(256) void tc_init_kernel(float* __restrict__ acc) {
  if (threadIdx.x < 8) acc[threadIdx.x] = 0.0f;
}

// One block (256 threads) per batch element.  ~66KB LDS.
__global__ __launch_bounds__(256) void tc_dp_kernel(
    const float* __restrict__ span_logits,
    const int* __restrict__ spans_ind,
    const unsigned char* __restrict__ maskspan,
    float* __restrict__ acc) {
  __shared__ float S1[STR * LDIM];  // [i*STR+j] i<=j: beta ; [j*STR+i] i<j: F
  __shared__ float FD[LDIM];        // F on the diagonal
  __shared__ int n_sh;
  const int b = blockIdx.x, tid = threadIdx.x;
  const float2* X2 = (const float2*)span_logits + (size_t)b * LDIM * LDIM;

  if (tid == 0) n_sh = 0;
  __syncthreads();
  if (tid < LDIM) {
    const unsigned char* mr = maskspan + (size_t)b * LDIM * LDIM;  // row i=0
    if (mr[tid]) atomicAdd(&n_sh, 1);
  }
  for (int idx = tid; idx < LDIM * LDIM; idx += 256) {
    int i = idx >> 7, j = idx & 127;
    float v = NEGV;
    if (i == j) {
      float2 x = X2[idx];
      v = lse2(x.x, x.y);  // scores(i,i) == beta(i,i)
    }
    S1[i * STR + j] = v;   // inits both triangles to NEG (diag = score)
  }
  __syncthreads();
  const int n = n_sh;

  // ---------------- inside (widths ascending) ----------------
  for (int w = 1; w <= n - 1; ++w) {
    const int cells = n - w;
    int rs = 0;                                   // R = 2^rs lanes per cell
    while (((2 << rs) * cells) <= 256 && rs < 5) rs++;
    const int R = 1 << rs;
    const int a = tid >> rs, sub = tid & (R - 1);
    float m = NEGV, s = 0.0f;
    if (a < cells) {
      const int j = a + w;
      for (int k = sub; k < w; k += R) {
        float t = S1[a * STR + a + k] + S1[(a + k + 1) * STR + j];
        lse_acc(m, s, t);
      }
    }
    for (int o = R >> 1; o >= 1; o >>= 1) {
      float om = __shfl_xor(m, o, 32);
      float os = __shfl_xor(s, o, 32);
      lse_combine(m, s, om, os);
    }
    if (a < cells && sub == 0) {
      float2 x = X2[a * LDIM + (a + w)];
      S1[a * STR + a + w] = lse2(x.x, x.y) + m + __logf(s);  // beta(a,a+w)
    }
    __syncthreads();
  }

  // ---------------- outside top cell ----------------
  if (tid == 0) {
    float2 x = X2[n - 1];                                // cell (0, n-1)
    S1[(n - 1) * STR + 0] = lse2(x.x, x.y) - S1[n - 1];  // F = sc - beta
  }
  __syncthreads();

  // ---------------- outside (widths descending, gather form) ----------------
  for (int w = n - 2; w >= 0; --w) {
    const int cells = n - w;
    int rs = 0;
    while (((2 << rs) * cells) <= 256 && rs < 5) rs++;
    const int R = 1 << rs;
    const int a = tid >> rs, sub = tid & (R - 1);
    float m = NEGV, s = 0.0f;
    const int bb = a + w;
    if (a < cells) {
      const int T1 = n - 1 - bb;  // parents (a, j) with j = bb+1+p
      const int T = T1 + a;       // then parents (i, bb) with i = p-T1
      for (int p = sub; p < T; p += R) {
        float t;
        if (p < T1) {
          int j = bb + 1 + p;
          t = S1[j * STR + a] + S1[(bb + 1) * STR + j];  // F(a,j)+beta(bb+1,j)
        } else {
          int i = p - T1;
          t = S1[bb * STR + i] + S1[i * STR + (a - 1)];  // F(i,bb)+beta(i,a-1)
        }
        lse_acc(m, s, t);
      }
    }
    for (int o = R >> 1; o >= 1; o >>= 1) {
      float om = __shfl_xor(m, o, 32);
      float os = __shfl_xor(s, o, 32);
      lse_combine(m, s, om, os);
    }
    if (a < cells && sub == 0) {
      float2 x = X2[a * LDIM + bb];
      float Fv = lse2(x.x, x.y) + m + __logf(s);  // F = sc + LSE(parents)
      if (a == bb) FD[a] = Fv;
      else S1[bb * STR + a] = Fv;
    }
    __syncthreads();
  }

  // ---------------- span loss accumulation ----------------
  float lsum = 0.0f;
  const int* SP = spans_ind + (size_t)b * LDIM * LDIM;
  for (int idx = tid; idx < LDIM * LDIM; idx += 256) {
    int i = idx >> 7, j = idx & 127;
    if (i <= j && j < n) {
      float Fv = (i == j) ? FD[i] : S1[j * STR + i];
      float beta = S1[i * STR + j];
      float2 x = X2[idx];
      float sc = lse2(x.x, x.y);
      // spans_b true iff spans_ind == 2 (where(ind>0, ind-1, ind) as bool)
      float xc = (SP[idx] == 2) ? x.y : x.x;
      float lm = Fv + beta - 2.0f * sc + xc;  // log marginal of chosen channel
      lsum += fmaxf(lm, LOGEPS);
    }
  }
  lsum = wave_sum(lsum);
  if ((tid & 31) == 0) atomicAdd(&acc[3], lsum);
  if (tid == 0) atomicAdd(&acc[4], (float)n);
}

// BCE over ph/pt with maskarc. Streams staged global->LDS via CDNA5 async
// copy; per-lane partial sums reduced with V_WMMA_F32_16X16X4_F32 (B = ones).
#define TOTV4 ((BDIM * LDIM * LDIM) / 4)  // 131072 float4s per array

__global__ __launch_bounds__(256) void tc_bce_kernel(
    const float* __restrict__ ph, const float* __restrict__ pt,
    const int* __restrict__ ph_ind, const int* __restrict__ pt_ind,
    const unsigned char* __restrict__ maskarc, float* __restrict__ acc) {
  __shared__ float stage_ph[256 * 4];
  __shared__ float stage_pt[256 * 4];
  const int tid = threadIdx.x;
  const int gl = blockIdx.x * blockDim.x + tid;
  const int stride = gridDim.x * blockDim.x;  // 65536 -> exactly 2 iterations
  // low 32 bits of a generic pointer into LDS == LDS byte offset
  unsigned lph = (unsigned)(unsigned long long)(uintptr_t)(&stage_ph[tid * 4]);
  unsigned lpt = (unsigned)(unsigned long long)(uintptr_t)(&stage_pt[tid * 4]);

#ifdef HAVE_WMMA_X4
  v8f cph = {0.f, 0.f, 0.f, 0.f, 0.f, 0.f, 0.f, 0.f};
  v8f cpt = {0.f, 0.f, 0.f, 0.f, 0.f, 0.f, 0.f, 0.f};
  v2f ones = {1.0f, 1.0f};
#else
  float sph_sc = 0.0f, spt_sc = 0.0f;
#endif
  float cnt = 0.0f;

  for (int base = gl; base < TOTV4; base += stride) {  // uniform trip count
    unsigned long long gph = (unsigned long long)(uintptr_t)(ph + (size_t)base * 4);
    unsigned long long gpt = (unsigned long long)(uintptr_t)(pt + (size_t)base * 4);
    asm volatile("global_load_async_to_lds_b128 %0, %1, off"
                 :: "v"(lph), "v"(gph) : "memory");
    asm volatile("global_load_async_to_lds_b128 %0, %1, off"
                 :: "v"(lpt), "v"(gpt) : "memory");
    asm volatile("s_wait_asynccnt 0" ::: "memory");

    float lh[4], lt[4];
#pragma unroll
    for (int u = 0; u < 4; ++u) {
      int idx = base * 4 + u;
      float mk = maskarc[idx] ? 1.0f : 0.0f;
      float xh = stage_ph[tid * 4 + u];
      float yh = (float)ph_ind[idx];
      lh[u] = mk * (fmaxf(xh, 0.0f) - xh * yh +
                    __logf(1.0f + __expf(-fabsf(xh))));
      float xt = stage_pt[tid * 4 + u];
      float yt = (float)pt_ind[idx];
      lt[u] = mk * (fmaxf(xt, 0.0f) - xt * yt +
                    __logf(1.0f + __expf(-fabsf(xt))));
      cnt += mk;
    }
    v2f aph = {lh[0] + lh[1], lh[2] + lh[3]};
    v2f apt = {lt[0] + lt[1], lt[2] + lt[3]};
#ifdef HAVE_WMMA_X4
    // D = A x ones + C : every D column holds the running row sums (exact f32)
    cph = __builtin_amdgcn_wmma_f32_16x16x4_f32(false, aph, false, ones,
                                                (short)0, cph, false, false);
    cpt = __builtin_amdgcn_wmma_f32_16x16x4_f32(false, apt, false, ones,
                                                (short)0, cpt, false, false);
#else
    sph_sc += aph.x + aph.y;
    spt_sc += apt.x + apt.y;
#endif
  }

  float sph, spt;
#ifdef HAVE_WMMA_X4
  // sum of all D elements == 16 * (sum of all A inputs)
  sph = (cph[0] + cph[1] + cph[2] + cph[3] + cph[4] + cph[5] + cph[6] + cph[7]) * 0.0625f;
  spt = (cpt[0] + cpt[1] + cpt[2] + cpt[3] + cpt[4] + cpt[5] + cpt[6] + cpt[7]) * 0.0625f;
#else
  sph = sph_sc;
  spt = spt_sc;
#endif
  sph = wave_sum(sph);
  spt = wave_sum(spt);
  cnt = wave_sum(cnt);
  if ((tid & 31) == 0) {
    atomicAdd(&acc[0], sph);
    atomicAdd(&acc[1], spt);
    atomicAdd(&acc[2], cnt);
  }
}

__global__ __launch_bounds__(32) void tc_fin_kernel(const float* __restrict__ acc,
                                                    float* __restrict__ out) {
  if (threadIdx.x == 0 && blockIdx.x == 0) {
    float loss_spans = -acc[3] / acc[4];
    float inv = 1.0f / acc[2];
    out[0] = 0.5f * loss_spans + 0.5f * (acc[0] * inv + acc[1] * inv);
  }
}

extern "C" void kernel_launch(void* const* d_in, const int* in_sizes, int n_in,
                              void* d_out, int out_size, void* d_ws, size_t ws_size,
                              hipStream_t stream) {
  const float* span_logits = (const float*)d_in[0];
  const float* ph = (const float*)d_in[1];
  const float* pt = (const float*)d_in[2];
  // d_in[3] = ph_arc : unused by the reference loss
  const int* spans_ind = (const int*)d_in[4];
  const int* ph_ind = (const int*)d_in[5];
  const int* pt_ind = (const int*)d_in[6];
  const unsigned char* maskspan = (const unsigned char*)d_in[7];
  const unsigned char* maskarc = (const unsigned char*)d_in[8];
  float* acc = (float*)d_ws;  // [ph_sum, pt_sum, arc_cnt, span_sum, lens_sum, ...]
  float* out = (float*)d_out;

  tc_init_kernel<<<1, 256, 0, stream>>>(acc);
  tc_dp_kernel<<<BDIM, 256, 0, stream>>>(span_logits, spans_ind, maskspan, acc);
  tc_bce_kernel<<<256, 256, 0, stream>>>(ph, pt, ph_ind, pt_ind, maskarc, acc);
  tc_fin_kernel<<<1, 32, 0, stream>>>(acc, out);
}